// GCN_28535762715201
// MI455X (gfx1250) — compile-verified
//
#include <hip/hip_runtime.h>
#include <math.h>

// ---------------------------------------------------------------------------
// GCN forward on MI455X (gfx1250): 2x GCNConv + log_softmax, all f32.
// GEMMs use V_WMMA_F32_16X16X4_F32 (native f32 WMMA, wave32) with
// LDS-staged B (pair-interleaved -> single ds_load_b64 per fragment) and
// 2 M-tiles per wave for A/B fragment reuse and latency hiding.
// ---------------------------------------------------------------------------

typedef __attribute__((ext_vector_type(2))) float v2f;
typedef __attribute__((ext_vector_type(8))) float v8f;

#define N_NODES 50000
#define IN_DIM  256
#define HID_DIM 64
#define OUT_DIM 40
#define OUT_PAD 48
#define N_EDGES 800000

// ---------------- degree / norm ----------------

__global__ void init_deg_kernel(float* __restrict__ deg, int n) {
    int i = blockIdx.x * blockDim.x + threadIdx.x;
    if (i < n) deg[i] = 1.0f;   // self-loop contributes 1
}

__global__ void deg_accum_kernel(const int* __restrict__ dst, float* __restrict__ deg, int e) {
    int i = blockIdx.x * blockDim.x + threadIdx.x;
    if (i < e) atomicAdd(&deg[dst[i]], 1.0f);
}

__global__ void dinv_kernel(float* __restrict__ deg, int n) {
    int i = blockIdx.x * blockDim.x + threadIdx.x;
    if (i < n) deg[i] = __frsqrt_rn(deg[i]);   // deg >= 1 always
}

// ---------------- generic zero ----------------

__global__ void zero_kernel(float* __restrict__ p, int n) {
    int i = blockIdx.x * blockDim.x + threadIdx.x;
    if (i < n) p[i] = 0.0f;
}

// ---------------- WMMA f32 GEMM: C[M,Nout] = A[M,K] @ B[K,Npad] ----------------
// Fragment layouts (wave32):
//   A (16x4 f32): lane l holds row M = l%16; VGPRs = K(2*(l/16)), K(2*(l/16)+1)
//                 -> contiguous float2 load from global per lane.
//   B (4x16 f32): lane l holds col N = l%16; VGPRs = K(2*(l/16)), K(2*(l/16)+1).
//                 B is staged in LDS pair-interleaved: ldsB[(kk/2)*Npad + n] =
//                 {B[kk][n], B[kk+1][n]} (kk even) -> one ds_load_b64/fragment.
//   C/D (16x16 f32): VGPR r, lanes 0-15: (M=r, N=lane); lanes 16-31: (M=r+8).
// Each wave computes MT=2 M-tiles x NT N-tiles; per K-step: MT global_load_b64
// + NT ds_load_b64 feed MT*NT back-to-back WMMAs.

template <int NT, int MT>
__global__ void gemm_wmma_kernel(const float* __restrict__ A,
                                 const float* __restrict__ B,
                                 float* __restrict__ C,
                                 int M, int K, int Npad, int Nout) {
    extern __shared__ v2f ldsB[];   // (K/2) * Npad fragments

    const int lane = threadIdx.x & 31;
    const int wave = threadIdx.x >> 5;
    const int wavesPerBlock = blockDim.x >> 5;
    const int lo  = lane & 15;
    const int hi  = lane >> 4;      // 0 or 1

    // ---- stage B into LDS, pair-interleaved along K ----
    {
        const int pairs = K >> 1;
        const int total = pairs * Npad;
        for (int i = threadIdx.x; i < total; i += blockDim.x) {
            int p = i / Npad;            // K-pair index
            v2f v;
            v.x = B[(size_t)i + (size_t)p * Npad];          // B[2p][c]
            v.y = B[(size_t)i + (size_t)(p + 1) * Npad];    // B[2p+1][c]
            ldsB[i] = v;
        }
    }
    __syncthreads();

    const int tile0 = (blockIdx.x * wavesPerBlock + wave) * MT;
    if (tile0 * 16 >= M) return;    // wave-uniform: EXEC stays all-1s below

    int  m0[MT];
    bool valid[MT];
    const float* __restrict__ arow[MT];
#pragma unroll
    for (int mt = 0; mt < MT; ++mt) {
        int t = tile0 + mt;
        valid[mt] = (t * 16 < M);
        m0[mt] = valid[mt] ? t * 16 : 0;     // clamp: compute-but-don't-store
        arow[mt] = A + (size_t)(m0[mt] + lo) * K;
    }

    v8f acc[MT][NT];
#pragma unroll
    for (int mt = 0; mt < MT; ++mt)
#pragma unroll
        for (int t = 0; t < NT; ++t)
#pragma unroll
            for (int j = 0; j < 8; ++j) acc[mt][t][j] = 0.0f;

    for (int k = 0; k < K; k += 4) {
        v2f a[MT];
#pragma unroll
        for (int mt = 0; mt < MT; ++mt)
            a[mt] = *(const v2f*)(arow[mt] + k + 2 * hi);

        v2f b[NT];
        const int pbase = ((k >> 1) + hi) * Npad + lo;
#pragma unroll
        for (int t = 0; t < NT; ++t)
            b[t] = ldsB[pbase + t * 16];

#pragma unroll
        for (int mt = 0; mt < MT; ++mt)
#pragma unroll
            for (int t = 0; t < NT; ++t)
                acc[mt][t] = __builtin_amdgcn_wmma_f32_16x16x4_f32(
                    false, a[mt], false, b[t],
                    (short)0, acc[mt][t], false, false);
    }

#pragma unroll
    for (int mt = 0; mt < MT; ++mt) {
        if (!valid[mt]) continue;
#pragma unroll
        for (int t = 0; t < NT; ++t) {
            const int col = t * 16 + lo;
            if (col < Nout) {
#pragma unroll
                for (int r = 0; r < 8; ++r) {
                    const int row = m0[mt] + r + 8 * hi;
                    C[(size_t)row * Nout + col] = acc[mt][t][r];
                }
            }
        }
    }
}

// ---------------- edge scatter (sym-normalized), incl. self-loops ----------------
// 64 consecutive threads handle one edge -> feature loads/stores are coalesced,
// edge-index loads broadcast from cache.

__global__ void scatter_kernel(const float* __restrict__ h,
                               const int* __restrict__ src,
                               const int* __restrict__ dst,
                               const float* __restrict__ dinv,
                               float* __restrict__ out,
                               int e, int nNodes, int F, int Fact) {
    int t = blockIdx.x * blockDim.x + threadIdx.x;
    int edge = t >> 6;
    int f = t & 63;
    if (edge >= e + nNodes) return;
    if (f >= Fact) return;
    int s, d;
    if (edge < e) { s = src[edge]; d = dst[edge]; }
    else          { s = edge - e;  d = s; }          // self-loop
    float w = dinv[s] * dinv[d];
    atomicAdd(out + (size_t)d * F + f, h[(size_t)s * F + f] * w);
}

// ---------------- bias + ReLU (F == 64, in place) ----------------

__global__ void bias_relu_kernel(float* __restrict__ h, const float* __restrict__ b, int n) {
    int i = blockIdx.x * blockDim.x + threadIdx.x;
    if (i < n) {
        float v = h[i] + b[i & 63];
        h[i] = v > 0.0f ? v : 0.0f;
    }
}

// ---------------- pad W2 [64,40] -> [64,48] with zeros ----------------

__global__ void pad_w2_kernel(const float* __restrict__ W2, float* __restrict__ W2p) {
    int i = blockIdx.x * blockDim.x + threadIdx.x;
    if (i < HID_DIM * OUT_PAD) {
        int r = i / OUT_PAD, c = i % OUT_PAD;
        W2p[i] = (c < OUT_DIM) ? W2[r * OUT_DIM + c] : 0.0f;
    }
}

// ---------------- fused bias + log_softmax over 40 cols, in place ----------------
// One wave32 per node; lane covers features {lane, lane+32}.

__global__ void bias_logsoftmax_kernel(float* __restrict__ out,
                                       const float* __restrict__ b2, int n) {
    const int lane = threadIdx.x & 31;
    const int wave = threadIdx.x >> 5;
    const int node = blockIdx.x * (blockDim.x >> 5) + wave;
    if (node >= n) return;
    float* __restrict__ row = out + (size_t)node * OUT_DIM;

    float v0 = row[lane] + b2[lane];                 // lane < 32 < 40: always valid
    float v1 = (lane < OUT_DIM - 32) ? (row[32 + lane] + b2[32 + lane])
                                     : -__builtin_inff();
    float m = fmaxf(v0, v1);
#pragma unroll
    for (int off = 16; off > 0; off >>= 1) m = fmaxf(m, __shfl_xor(m, off, 32));
    float s = expf(v0 - m) + ((lane < OUT_DIM - 32) ? expf(v1 - m) : 0.0f);
#pragma unroll
    for (int off = 16; off > 0; off >>= 1) s += __shfl_xor(s, off, 32);
    float ls = m + logf(s);
    row[lane] = v0 - ls;
    if (lane < OUT_DIM - 32) row[32 + lane] = v1 - ls;
}

// ---------------------------------------------------------------------------

static inline int cdiv(int a, int b) { return (a + b - 1) / b; }

extern "C" void kernel_launch(void* const* d_in, const int* in_sizes, int n_in,
                              void* d_out, int out_size, void* d_ws, size_t ws_size,
                              hipStream_t stream) {
    const float* x  = (const float*)d_in[0];
    const int*   ei = (const int*)d_in[1];
    const float* W1 = (const float*)d_in[2];
    const float* b1 = (const float*)d_in[3];
    const float* W2 = (const float*)d_in[4];
    const float* b2 = (const float*)d_in[5];
    float* out = (float*)d_out;

    const int* src = ei;             // edge_index[0]
    const int* dst = ei + N_EDGES;   // edge_index[1]

    // Workspace layout (floats):
    float* ws   = (float*)d_ws;
    float* dinv = ws;                                   // 50000 (deg -> rsqrt in place)
    float* h1   = dinv + N_NODES;                       // 50000*64 ; reused as h2 (stride 40)
    float* agg1 = h1 + (size_t)N_NODES * HID_DIM;       // 50000*64
    float* w2p  = agg1 + (size_t)N_NODES * HID_DIM;     // 64*48

    const int nh = N_NODES * HID_DIM;      // 3,200,000
    const int no = N_NODES * OUT_DIM;      // 2,000,000
    const int tiles = N_NODES / 16;        // 3125 (exact)
    const int gemmBlocks = cdiv(tiles, 8 * 2);          // 8 waves * MT=2 tiles per block
    const size_t ldsB1 = (size_t)(IN_DIM / 2) * HID_DIM * sizeof(v2f);   // 64 KB
    const size_t ldsB2 = (size_t)(HID_DIM / 2) * OUT_PAD * sizeof(v2f);  // 12 KB
    const int scatterBlocks = cdiv((N_EDGES + N_NODES) * 64, 256);

    // 1) degrees with self-loops -> dinv
    init_deg_kernel<<<cdiv(N_NODES, 256), 256, 0, stream>>>(dinv, N_NODES);
    deg_accum_kernel<<<cdiv(N_EDGES, 256), 256, 0, stream>>>(dst, dinv, N_EDGES);
    dinv_kernel<<<cdiv(N_NODES, 256), 256, 0, stream>>>(dinv, N_NODES);

    // 2) h1 = x @ W1   (WMMA f32, 4 N-tiles, LDS-staged W1)
    gemm_wmma_kernel<4, 2><<<gemmBlocks, 256, ldsB1, stream>>>(
        x, W1, h1, N_NODES, IN_DIM, HID_DIM, HID_DIM);

    // 3) agg1 = scatter_add(norm * h1[src]) over edges + self-loops
    zero_kernel<<<cdiv(nh, 256), 256, 0, stream>>>(agg1, nh);
    scatter_kernel<<<scatterBlocks, 256, 0, stream>>>(h1, src, dst, dinv, agg1,
                                                      N_EDGES, N_NODES, HID_DIM, HID_DIM);

    // 4) agg1 = relu(agg1 + b1)
    bias_relu_kernel<<<cdiv(nh, 256), 256, 0, stream>>>(agg1, b1, nh);

    // 5) h2 = agg1 @ W2 (padded to 48 cols; only 40 stored, stride 40, into h1 buffer)
    pad_w2_kernel<<<cdiv(HID_DIM * OUT_PAD, 256), 256, 0, stream>>>(W2, w2p);
    gemm_wmma_kernel<3, 2><<<gemmBlocks, 256, ldsB2, stream>>>(
        agg1, w2p, h1, N_NODES, HID_DIM, OUT_PAD, OUT_DIM);

    // 6) out = scatter_add(norm * h2[src])
    zero_kernel<<<cdiv(no, 256), 256, 0, stream>>>(out, no);
    scatter_kernel<<<scatterBlocks, 256, 0, stream>>>(h1, src, dst, dinv, out,
                                                      N_EDGES, N_NODES, OUT_DIM, OUT_DIM);

    // 7) out = log_softmax(out + b2), one wave32 per node
    bias_logsoftmax_kernel<<<cdiv(N_NODES, 4), 128, 0, stream>>>(out, b2, N_NODES);
}